// FeatureMap_16381005266987
// MI455X (gfx1250) — compile-verified
//
#include <hip/hip_runtime.h>
#include <math.h>

typedef __attribute__((ext_vector_type(2))) float v2f;
typedef __attribute__((ext_vector_type(8))) float v8f;

#define BH        32      // b*h
#define SEQ       4096
#define DDIM      64
#define MDIM      256
#define NT        16      // 256 / 16 column tiles
#define KK        16      // 64 / 4 k-steps
#define PT_STRIDE 272     // 256 + 16 pad: half-wave bank ranges disjoint mod 64

__device__ __forceinline__ unsigned f2ord(float f) {
    unsigned u = __float_as_uint(f);
    return (u & 0x80000000u) ? ~u : (u | 0x80000000u);
}
__device__ __forceinline__ float ord2f(unsigned k) {
    unsigned u = (k & 0x80000000u) ? (k ^ 0x80000000u) : ~k;
    return __uint_as_float(u);
}

// MODE 0: query  -> per-row max stab, write out
// MODE 1: key p1 -> GEMM, block max, atomicMax into ws[bh] (no output)
// MODE 2: key p2 -> GEMM, stab = ws[bh], write out
template <int MODE>
__global__ __launch_bounds__(128) void performer_phi(
    const float* __restrict__ data,  // [BH*SEQ, DDIM]
    const float* __restrict__ proj,  // [MDIM, DDIM]
    float* __restrict__ out,         // [BH*SEQ, MDIM]
    unsigned* __restrict__ wsmax)    // [BH] order-encoded max
{
    __shared__ float ldsPT[DDIM * PT_STRIDE];  // PT[k][n] = proj[n][k]

    const int tid  = threadIdx.x;
    const int lane = tid & 31;
    const int wave = tid >> 5;
    const int half = lane >> 4;
    const int l16  = lane & 15;

    // ---- stage projection transposed into LDS (conflict-free stores) ----
    // float4 idx = i*16 + j4 ; assign consecutive tids consecutive i (rows of P)
    #pragma unroll
    for (int c = 0; c < 32; ++c) {
        int idx = c * 128 + tid;        // 0..4095
        int i   = idx & 255;            // n  (row of P)
        int j4  = idx >> 8;             // k/4
        float4 v = ((const float4*)proj)[i * 16 + j4];
        int j = j4 << 2;
        ldsPT[(j + 0) * PT_STRIDE + i] = v.x;
        ldsPT[(j + 1) * PT_STRIDE + i] = v.y;
        ldsPT[(j + 2) * PT_STRIDE + i] = v.z;
        ldsPT[(j + 3) * PT_STRIDE + i] = v.w;
    }
    __syncthreads();

    const size_t rowBlock = (size_t)blockIdx.x * 64 + wave * 16;
    const int bh = (int)(rowBlock / SEQ);
    const float dn = 0.35355339059327373f;  // 64^-0.25

    // ---- A fragments (16x4 f32 layout: lane = m%16, k = 4kk + 2*half + v) ----
    v2f afrag[KK];
    float ss = 0.f;  // sum of (dn*x)^2 for row l16
    {
        const float2* rowp = (const float2*)(data + (rowBlock + l16) * DDIM);
        #pragma unroll
        for (int kk = 0; kk < KK; ++kk) {
            float2 t = rowp[2 * kk + half];
            v2f a;
            a.x = t.x * dn;
            a.y = t.y * dn;
            afrag[kk] = a;
            ss += a.x * a.x + a.y * a.y;
        }
    }
    if (MODE != 1) ss += __shfl_xor(ss, 16, 32);  // combine both k-halves of the row

    // ---- GEMM: D[16x16] += A[16x4] x B[4x16], 16 col tiles x 16 k-steps ----
    v8f acc[NT];
    #pragma unroll
    for (int nt = 0; nt < NT; ++nt) acc[nt] = {};

    #pragma unroll
    for (int nt = 0; nt < NT; ++nt) {
        const int ncol = nt * 16 + l16;
        #pragma unroll
        for (int kk = 0; kk < KK; ++kk) {
            const int krow = 4 * kk + 2 * half;
            v2f b;
            b.x = ldsPT[(krow + 0) * PT_STRIDE + ncol];
            b.y = ldsPT[(krow + 1) * PT_STRIDE + ncol];
            acc[nt] = __builtin_amdgcn_wmma_f32_16x16x4_f32(
                false, afrag[kk], false, b, (short)0, acc[nt], false, false);
        }
    }

    if (MODE == 1) {
        // ---- block max of data_dash -> ordered-uint atomicMax per (b,h) ----
        float pm = -3.402823466e38f;
        #pragma unroll
        for (int nt = 0; nt < NT; ++nt)
            #pragma unroll
            for (int e = 0; e < 8; ++e) pm = fmaxf(pm, acc[nt][e]);
        #pragma unroll
        for (int m = 1; m <= 16; m <<= 1) pm = fmaxf(pm, __shfl_xor(pm, m, 32));
        if (lane == 0) atomicMax(&wsmax[bh], f2ord(pm));
        return;
    }

    // ---- epilogue: stab, diag, exp, store ----
    float stab_s = 0.f;
    if (MODE == 2) stab_s = ord2f(wsmax[bh]);

    float rmax[8];
    if (MODE == 0) {
        #pragma unroll
        for (int r = 0; r < 8; ++r) {
            float v = acc[0][r];
            #pragma unroll
            for (int nt = 1; nt < NT; ++nt) v = fmaxf(v, acc[nt][r]);
            // reduce across the 16 lanes of this half (rows r+8*half)
            #pragma unroll
            for (int m = 1; m <= 8; m <<= 1) v = fmaxf(v, __shfl_xor(v, m, 32));
            rmax[r] = v;
        }
    }

    const float ratio = 0.0625f;  // 256^-0.5
    #pragma unroll
    for (int r = 0; r < 8; ++r) {
        const int lrow = r + 8 * half;                 // D layout: VGPR r <-> row r+8*half
        const float diag = 0.5f * __shfl(ss, lrow, 32);
        const float st = (MODE == 0) ? rmax[r] : stab_s;
        float* op = out + (rowBlock + lrow) * MDIM;
        #pragma unroll
        for (int nt = 0; nt < NT; ++nt) {
            float v = ratio * (__expf(acc[nt][r] - diag - st) + 1e-4f);
            op[nt * 16 + l16] = v;
        }
    }
}

extern "C" void kernel_launch(void* const* d_in, const int* in_sizes, int n_in,
                              void* d_out, int out_size, void* d_ws, size_t ws_size,
                              hipStream_t stream) {
    const float* q    = (const float*)d_in[0];
    const float* k    = (const float*)d_in[1];
    const float* proj = (const float*)d_in[2];
    float* out        = (float*)d_out;
    unsigned* ws      = (unsigned*)d_ws;

    const size_t N = (size_t)BH * SEQ * MDIM;  // per-tensor output elements

    hipMemsetAsync(ws, 0, BH * sizeof(unsigned), stream);  // 0 < ordered(-inf): identity for atomicMax

    dim3 grid(BH * SEQ / 64), block(128);
    performer_phi<0><<<grid, block, 0, stream>>>(q, proj, out, ws);      // q_prime
    performer_phi<1><<<grid, block, 0, stream>>>(k, proj, out, ws);      // k max pass
    performer_phi<2><<<grid, block, 0, stream>>>(k, proj, out + N, ws);  // k_prime
}